// NodeClassifier_8375186227359
// MI455X (gfx1250) — compile-verified
//
#include <hip/hip_runtime.h>

// Problem constants (fixed by the reference; E taken from in_sizes[0]).
#define NNODES 20000
#define NRELH  16                 // nrel
#define RTOT   33                 // 2*nrel + 1
#define HID    64
#define NCLS   16
#define RN     (RTOT * NNODES)    // 660000 segments
#define PCOLS  (RTOT * NCLS)      // 528

typedef __attribute__((ext_vector_type(2))) float v2f;
typedef __attribute__((ext_vector_type(8))) float v8f;

// ---------------- zero scratch ----------------
__global__ __launch_bounds__(256) void k_zero(float* __restrict__ p, int n) {
  int i = blockIdx.x * blockDim.x + threadIdx.x;
  if (i < n) p[i] = 0.0f;
}

// ---------------- segment counts (orig + inverse blocks; self-loops are analytically 1) ----------------
__global__ __launch_bounds__(256) void k_count(const int* __restrict__ src,
                                               const int* __restrict__ rel,
                                               const int* __restrict__ dst,
                                               float* __restrict__ cnt_ro,
                                               float* __restrict__ cnt_rs, int E) {
  int e = blockIdx.x * blockDim.x + threadIdx.x;
  if (e >= E) return;
  int s = src[e], r = rel[e], d = dst[e];
  atomicAdd(cnt_ro + (size_t)r * NNODES + d, 1.0f);                 // ro of original
  atomicAdd(cnt_ro + (size_t)(r + NRELH) * NNODES + s, 1.0f);      // ro of inverse
  atomicAdd(cnt_rs + (size_t)r * NNODES + s, 1.0f);                 // rs of original
  atomicAdd(cnt_rs + (size_t)(r + NRELH) * NNODES + d, 1.0f);      // rs of inverse
}

// ---------------- layer 1: gather w1 row, scale by swapped 1/count, scatter to h ----------------
__global__ __launch_bounds__(256) void k_layer1(const int* __restrict__ src,
                                                const int* __restrict__ rel,
                                                const int* __restrict__ dst,
                                                const float* __restrict__ w1,
                                                const float* __restrict__ cnt_ro,
                                                float* __restrict__ hacc, int E) {
  long idx = (long)blockIdx.x * blockDim.x + threadIdx.x;
  if (idx >= (long)E * 16) return;
  int e  = (int)(idx >> 4);
  int i0 = ((int)idx & 15) * 4;
  int s = src[e], r = rel[e], d = dst[e];
  // swapped normalization (torch-rgcn): orig edge uses inverse's (r,o) count and vice versa
  float vo = 1.0f / cnt_ro[(size_t)(r + NRELH) * NNODES + s];
  float vi = 1.0f / cnt_ro[(size_t)r * NNODES + d];
  float4 a = *(const float4*)(w1 + ((size_t)r * NNODES + d) * HID + i0);
  atomicAdd(&hacc[(size_t)s * HID + i0 + 0], vo * a.x);
  atomicAdd(&hacc[(size_t)s * HID + i0 + 1], vo * a.y);
  atomicAdd(&hacc[(size_t)s * HID + i0 + 2], vo * a.z);
  atomicAdd(&hacc[(size_t)s * HID + i0 + 3], vo * a.w);
  float4 b = *(const float4*)(w1 + ((size_t)(r + NRELH) * NNODES + s) * HID + i0);
  atomicAdd(&hacc[(size_t)d * HID + i0 + 0], vi * b.x);
  atomicAdd(&hacc[(size_t)d * HID + i0 + 1], vi * b.y);
  atomicAdd(&hacc[(size_t)d * HID + i0 + 2], vi * b.z);
  atomicAdd(&hacc[(size_t)d * HID + i0 + 3], vi * b.w);
}

// ---------------- h finalize: + self-loop message (count==1) + b1, ReLU ----------------
__global__ __launch_bounds__(256) void k_hfinal(const float* __restrict__ w1,
                                                const float* __restrict__ b1,
                                                float* __restrict__ h) {
  int idx = blockIdx.x * blockDim.x + threadIdx.x;
  if (idx >= NNODES * HID) return;
  int n = idx / HID, i = idx % HID;
  float v = h[idx] + w1[((size_t)(2 * NRELH) * NNODES + n) * HID + i] + b1[i];
  h[idx] = v > 0.0f ? v : 0.0f;
}

// ---------------- P = h (20000x64) x W2cat (64x528) via V_WMMA_F32_16X16X4_F32 ----------------
// One wave per (16-node tile, relation). K=64 in steps of 4 -> 16 WMMA ops.
// A layout (f32 16x4): lane L holds A[M=L%16, K = 2*(L/16)+j] in VGPR j.
// B layout (f32 4x16): lane L holds B[K = 2*(L/16)+j, N=L%16] in VGPR j.
// C/D layout: VGPR v -> M = v + 8*(L/16), N = L%16.
__global__ __launch_bounds__(32) void k_gemm(const float* __restrict__ h,
                                             const float* __restrict__ w2,
                                             float* __restrict__ P) {
  int tile = blockIdx.x;
  int mt = tile % (NNODES / 16);       // 1250 node tiles
  int r  = tile / (NNODES / 16);       // 33 relations
  int lane  = threadIdx.x;
  int mn    = lane & 15;               // M row (for A) / N col (for B,C,D)
  int khalf = lane >> 4;               // K pair selector
  const float* hrow = h + (size_t)(mt * 16 + mn) * HID;
  const float* wrel = w2 + (size_t)r * HID * NCLS;   // 64x16 row-major
  union { v8f v; float f[8]; } acc;
  acc.v = (v8f)0.0f;
#pragma unroll
  for (int k0 = 0; k0 < HID; k0 += 4) {
    int ka = k0 + 2 * khalf;
    v2f a, b;
    a.x = hrow[ka];
    a.y = hrow[ka + 1];
    b.x = wrel[(size_t)ka * NCLS + mn];
    b.y = wrel[(size_t)(ka + 1) * NCLS + mn];
    acc.v = __builtin_amdgcn_wmma_f32_16x16x4_f32(
        /*neg_a=*/false, a, /*neg_b=*/false, b,
        /*c_mod=*/(short)0, acc.v, /*reuse_a=*/false, /*reuse_b=*/false);
  }
  int rowoff = khalf * 8;
#pragma unroll
  for (int v = 0; v < 8; ++v) {
    P[(size_t)(mt * 16 + v + rowoff) * PCOLS + r * NCLS + mn] = acc.f[v];
  }
}

// ---------------- out init: b2 + self-loop layer-2 term (vals2 == 1) ----------------
__global__ __launch_bounds__(256) void k_outinit(const float* __restrict__ b2,
                                                 const float* __restrict__ P,
                                                 float* __restrict__ out) {
  int idx = blockIdx.x * blockDim.x + threadIdx.x;
  if (idx >= NNODES * NCLS) return;
  int n = idx / NCLS, c = idx % NCLS;
  out[idx] = b2[c] + P[(size_t)n * PCOLS + (2 * NRELH) * NCLS + c];
}

// ---------------- layer 2: out[s] += P[o, r*16+c] / cnt_rs[rs] for both edge directions ----------------
__global__ __launch_bounds__(256) void k_layer2(const int* __restrict__ src,
                                                const int* __restrict__ rel,
                                                const int* __restrict__ dst,
                                                const float* __restrict__ cnt_rs,
                                                const float* __restrict__ P,
                                                float* __restrict__ out, int E) {
  long idx = (long)blockIdx.x * blockDim.x + threadIdx.x;
  if (idx >= (long)E * 16) return;
  int e = (int)(idx >> 4);
  int c = (int)idx & 15;
  int s = src[e], r = rel[e], d = dst[e];
  float vo = 1.0f / cnt_rs[(size_t)r * NNODES + s];
  float vi = 1.0f / cnt_rs[(size_t)(r + NRELH) * NNODES + d];
  atomicAdd(&out[(size_t)s * NCLS + c], vo * P[(size_t)d * PCOLS + r * NCLS + c]);
  atomicAdd(&out[(size_t)d * NCLS + c], vi * P[(size_t)s * PCOLS + (r + NRELH) * NCLS + c]);
}

extern "C" void kernel_launch(void* const* d_in, const int* in_sizes, int n_in,
                              void* d_out, int out_size, void* d_ws, size_t ws_size,
                              hipStream_t stream) {
  const int*   src = (const int*)d_in[0];
  const int*   rel = (const int*)d_in[1];
  const int*   dst = (const int*)d_in[2];
  const float* w1  = (const float*)d_in[3];
  const float* b1  = (const float*)d_in[4];
  const float* w2  = (const float*)d_in[5];
  const float* b2  = (const float*)d_in[6];
  float* out = (float*)d_out;
  const int E = in_sizes[0];

  // workspace carve-up (floats): cnt_ro | cnt_rs | h | P  => ~52.6 MB total
  float* cnt_ro = (float*)d_ws;
  float* cnt_rs = cnt_ro + RN;
  float* h      = cnt_rs + RN;
  float* P      = h + (size_t)NNODES * HID;

  const int zn = 2 * RN + NNODES * HID;
  k_zero<<<(zn + 255) / 256, 256, 0, stream>>>(cnt_ro, zn);

  k_count<<<(E + 255) / 256, 256, 0, stream>>>(src, rel, dst, cnt_ro, cnt_rs, E);

  long l1 = (long)E * 16;
  k_layer1<<<(unsigned)((l1 + 255) / 256), 256, 0, stream>>>(src, rel, dst, w1, cnt_ro, h, E);

  k_hfinal<<<(NNODES * HID + 255) / 256, 256, 0, stream>>>(w1, b1, h);

  k_gemm<<<(NNODES / 16) * RTOT, 32, 0, stream>>>(h, w2, P);

  k_outinit<<<(NNODES * NCLS + 255) / 256, 256, 0, stream>>>(b2, P, out);

  long l2 = (long)E * 16;
  k_layer2<<<(unsigned)((l2 + 255) / 256), 256, 0, stream>>>(src, rel, dst, cnt_rs, P, out, E);
}